// TypeAttention_9302899163641
// MI455X (gfx1250) — compile-verified
//
#include <hip/hip_runtime.h>
#include <hip/hip_bf16.h>

typedef float v2f __attribute__((ext_vector_type(2)));
typedef float v4f __attribute__((ext_vector_type(4)));
typedef float v8f __attribute__((ext_vector_type(8)));

#define Bdim 2
#define Sdim 128
#define Hdim 256
#define Cdim 8
#define Mdim (Bdim * Sdim)   // 256 rows of z
#define HQ   (Hdim / 4)      // row length in float4 quads

// ---------------------------------------------------------------------------
// K1: z[m,n] = relu( sum_k mem[m,k] * W_w[n,k] + W_b[n] )
// M=N=K=256, fp32 WMMA 16x16x4. One wave per 16x16 tile, 64 k-steps.
// (confirmed: lowers to v_wmma_f32_16x16x4_f32 with pipelined b64 loads)
// ---------------------------------------------------------------------------
__global__ __launch_bounds__(128) void ta_gemm_z(
    const float* __restrict__ mem, const float* __restrict__ Ww,
    const float* __restrict__ Wb, float* __restrict__ z) {
  const int lane  = threadIdx.x;        // 0..31 (wave32)
  const int row16 = lane & 15;          // M index for A, N index for B/C/D
  const int khalf = lane >> 4;          // which K pair this half-wave owns
  const int tm = blockIdx.x * 16;
  const int tn = (blockIdx.y * blockDim.y + threadIdx.y) * 16;

  const float* Arow = mem + (size_t)(tm + row16) * Hdim;  // A[m,k] = mem[m,k]
  const float* Brow = Ww  + (size_t)(tn + row16) * Hdim;  // B[k,n] = Ww[n,k]

  v8f acc = {};
  for (int k0 = 0; k0 < Hdim; k0 += 4) {
    const int k = k0 + 2 * khalf;
    v2f a = { Arow[k], Arow[k + 1] };
    v2f b = { Brow[k], Brow[k + 1] };
    acc = __builtin_amdgcn_wmma_f32_16x16x4_f32(
        /*neg_a=*/false, a, /*neg_b=*/false, b,
        /*c_mod=*/(short)0, acc, /*reuse_a=*/false, /*reuse_b=*/false);
  }

  // C/D layout: lane&15 = N, VGPR r -> M = r + 8*(lane>>4)
  const float bias = Wb[tn + row16];
#pragma unroll
  for (int r = 0; r < 8; ++r) {
    const int row = tm + r + 8 * khalf;
    const float v = acc[r] + bias;
    z[(size_t)row * Hdim + tn + row16] = v > 0.f ? v : 0.f;
  }
}

// ---------------------------------------------------------------------------
// K2a: score[b,c,s] = sum_h U_w[c,h]*z[b,s,h]; max over s; exp; cumsum over s.
// One block per (b,c), 128 threads (one per s). Negligible work.
// ---------------------------------------------------------------------------
__global__ __launch_bounds__(128) void ta_score_scan(
    const float* __restrict__ z, const float* __restrict__ Uw,
    float* __restrict__ exp_score, float* __restrict__ cum_exp) {
  const int b = blockIdx.x / Cdim;
  const int c = blockIdx.x % Cdim;
  const int s = threadIdx.x;
  __shared__ float sm[Sdim];

  const float* zp = z + (size_t)(b * Sdim + s) * Hdim;
  const float* up = Uw + c * Hdim;
  float acc = 0.f;
  for (int h = 0; h < Hdim; ++h) acc = fmaf(up[h], zp[h], acc);

  sm[s] = acc;
  __syncthreads();
  for (int off = 64; off >= 1; off >>= 1) {
    if (s < off) sm[s] = fmaxf(sm[s], sm[s + off]);
    __syncthreads();
  }
  const float mx = sm[0];
  __syncthreads();

  const float e = expf(acc - mx);
  exp_score[(b * Cdim + c) * Sdim + s] = e;
  sm[s] = e;
  __syncthreads();
  if (s == 0) {
    float run = 0.f;
    for (int t = 0; t < Sdim; ++t) { run += sm[t]; sm[t] = run; }
  }
  __syncthreads();
  cum_exp[(b * Cdim + c) * Sdim + s] = sm[s];
}

// ---------------------------------------------------------------------------
// K2b: cum_m[b,c,s,h] = cumsum_s( mem[b,s,h] * exp_score[b,c,s] )
// One block per (b,c), 256 threads (one per h), coalesced over h.
// ---------------------------------------------------------------------------
__global__ __launch_bounds__(256) void ta_cum_m(
    const float* __restrict__ mem, const float* __restrict__ exp_score,
    float* __restrict__ cum_m) {
  const int h = threadIdx.x;
  const int b = blockIdx.x / Cdim;
  const int c = blockIdx.x % Cdim;
  const float* es = exp_score + (b * Cdim + c) * Sdim;
  float* cm = cum_m + (size_t)(b * Cdim + c) * Sdim * Hdim;
  float run = 0.f;
  for (int s = 0; s < Sdim; ++s) {
    run = fmaf(mem[(size_t)(b * Sdim + s) * Hdim + h], es[s], run);
    cm[(size_t)s * Hdim + h] = run;
  }
}

// ---------------------------------------------------------------------------
// K4: out[b,i,j,c] = (sum_h relu(cum_m[j,h]-cum_m[i-1,h]) * V_w[c,h]) / cc_exp
//                    + V_b[c]
// One wave per (b,c,i). Lane owns h = 8*lane + t (t=0..7): the wave still
// covers a contiguous 1 KB row (fully coalesced) but each lane's 8 floats are
// contiguous -> two global_load_b128 per row instead of eight b32.
// Li and V_w[c,:] stay in registers across all 128 j's; cc_exp > 0 after the
// <=1e-6 -> 100 replacement so the divide hoists out of the h-loop.
// Next j-row is prefetched (global_prefetch_b8) to hide L2 latency.
// ---------------------------------------------------------------------------
__global__ __launch_bounds__(256) void ta_span_out(
    const float* __restrict__ cum_m, const float* __restrict__ cum_exp,
    const float* __restrict__ Vw, const float* __restrict__ Vb,
    float* __restrict__ out) {
  const int lane = threadIdx.x & 31;
  const int wid  = blockIdx.x * 8 + (threadIdx.x >> 5);   // 2048 waves total
  const int i = wid % Sdim;
  const int c = (wid / Sdim) % Cdim;
  const int b = wid / (Sdim * Cdim);
  const int bc = b * Cdim + c;

  const v4f* Lbase = (const v4f*)(cum_m + (size_t)bc * Sdim * Hdim);
  const int q = lane * 2;                      // quad index within a row

  const v4f* vwp = (const v4f*)(Vw + (size_t)c * Hdim);
  const v4f vw0 = vwp[q], vw1 = vwp[q + 1];

  v4f Li0 = {}, Li1 = {};
  float ce_i = 0.f;
  if (i > 0) {
    const v4f* Lp = Lbase + (size_t)(i - 1) * HQ;
    Li0 = Lp[q];
    Li1 = Lp[q + 1];
    ce_i = cum_exp[bc * Sdim + (i - 1)];
  }
  const float vb = Vb[c];

  for (int j = 0; j < Sdim; ++j) {
    const v4f* Lj = Lbase + (size_t)j * HQ;
    const v4f a0 = Lj[q];
    const v4f a1 = Lj[q + 1];
    if (j + 1 < Sdim)  // prefetch next row chunk this lane will read
      __builtin_prefetch((const void*)(Lj + HQ + q), 0, 3);

    float ce = cum_exp[bc * Sdim + j] - ce_i;
    if (ce <= 1e-6f) ce = 100.f;
    const float inv = 1.0f / ce;

    float acc = 0.f;
#pragma unroll
    for (int t = 0; t < 4; ++t) {
      acc = fmaf(fmaxf(a0[t] - Li0[t], 0.f), vw0[t], acc);
      acc = fmaf(fmaxf(a1[t] - Li1[t], 0.f), vw1[t], acc);
    }
#pragma unroll
    for (int off = 16; off >= 1; off >>= 1) acc += __shfl_xor(acc, off, 32);

    if (lane == 0)
      out[(((size_t)b * Sdim + i) * Sdim + j) * Cdim + c] = acc * inv + vb;
  }
}

// ---------------------------------------------------------------------------
extern "C" void kernel_launch(void* const* d_in, const int* in_sizes, int n_in,
                              void* d_out, int out_size, void* d_ws, size_t ws_size,
                              hipStream_t stream) {
  const float* mem = (const float*)d_in[0];  // (B,S,H)
  const float* Ww  = (const float*)d_in[1];  // (H,H)
  const float* Wb  = (const float*)d_in[2];  // (H,)
  const float* Uw  = (const float*)d_in[3];  // (C,H)
  const float* Vw  = (const float*)d_in[4];  // (C,H)
  const float* Vb  = (const float*)d_in[5];  // (C,)
  float* out = (float*)d_out;                // (B,S,S,C)

  // workspace layout (floats)
  float* ws = (float*)d_ws;
  float* z         = ws;                                   // B*S*H   = 65536
  float* exp_score = z + (size_t)Bdim * Sdim * Hdim;       // B*C*S   = 2048
  float* cum_exp   = exp_score + (size_t)Bdim * Cdim * Sdim;
  float* cum_m     = cum_exp + (size_t)Bdim * Cdim * Sdim; // B*C*S*H = 524288

  // K1: z GEMM (WMMA). Grid: 16 M-tiles x (4 blocks * 4 waves) N-tiles.
  ta_gemm_z<<<dim3(Mdim / 16, 4), dim3(32, 4), 0, stream>>>(mem, Ww, Wb, z);

  // K2a: scores + max + exp + cumsum per (b,c)
  ta_score_scan<<<Bdim * Cdim, Sdim, 0, stream>>>(z, Uw, exp_score, cum_exp);

  // K2b: weighted-memory prefix sums
  ta_cum_m<<<Bdim * Cdim, Hdim, 0, stream>>>(mem, exp_score, cum_m);

  // K4: span dot-products. 2048 waves = B*C*S, 8 waves/block.
  ta_span_out<<<(Bdim * Cdim * Sdim) / 8, 256, 0, stream>>>(cum_m, cum_exp, Vw,
                                                            Vb, out);
}